// AtomRepresentationModel_39170101739766
// MI455X (gfx1250) — compile-verified
//
#include <hip/hip_runtime.h>
#include <hip/hip_bf16.h>
#include <math.h>

#define B_ 16
#define N_ 1000
#define E_ 25000
#define H_ 128
#define BN_ (B_ * N_)   // 16000
#define BE_ (B_ * E_)   // 400000
#define EDGE_REAL 50
#define EDGE_PAD 64
#define LOG2F_ 0.6931471805599453f

typedef float v2f __attribute__((ext_vector_type(2)));
typedef float v8f __attribute__((ext_vector_type(8)));

// ---------------------------------------------------------------------------
// CDNA5 fp32 WMMA: D(16x16) = A(16x4) * B(4x16) + C   (wave32, f32 precise)
// A frag: lane L holds M=L&15; VGPR0/1 = K {0,1} (lanes 0-15) or {2,3} (16-31)
// B frag: lane L holds N=L&15; VGPR0/1 = K {0,1} (lanes 0-15) or {2,3} (16-31)
// C/D   : VGPR r -> M = r (lanes 0-15) / 8+r (lanes 16-31), N = L&15
//
// Weights are pre-packed so a B fragment is ONE b64 load:
//   Wp[(k>>1)*2H + 2n] = { W[k][n], W[k+1][n] }   (zero-padded rows baked in)
// ---------------------------------------------------------------------------
__device__ __forceinline__ v8f wmma4(v2f a, v2f b, v8f c) {
  return __builtin_amdgcn_wmma_f32_16x16x4_f32(false, a, false, b, (short)0, c,
                                               false, false);
}

// fast shifted softplus: log(1+e^x) - log(2); arg of log is in [1,2] so the
// hardware v_log path stays well-conditioned
__device__ __forceinline__ float sspf(float x) {
  return fmaxf(x, 0.f) + __logf(1.f + __expf(-fabsf(x))) - LOG2F_;
}

// XOR-swizzled LDS index for a 16 x 128 f32 tile (keeps v2f pairs contiguous,
// conflict-free for both row-broadcast stores and column reads)
__device__ __forceinline__ int ldsIdx(int m, int k) {
  return m * H_ + (k ^ ((m & 15) << 1));
}

__device__ __forceinline__ void bias_init(v8f acc[8], const float* __restrict__ bias, int nl) {
#pragma unroll
  for (int nt = 0; nt < 8; ++nt) {
    float bv = bias[nt * 16 + nl];
    v8f c;
#pragma unroll
    for (int i = 0; i < 8; ++i) c[i] = bv;
    acc[nt] = c;
  }
}

// acc(16x128) += A_glb_rows(16x128) @ Wp(packed 128x128)
__device__ __forceinline__ void gemm_from_global(const float* __restrict__ a0,
                                                 const float* __restrict__ Wp,
                                                 int lane, v8f acc[8]) {
  const int nl = lane & 15;
  const int kofs = (lane >> 4) << 1;
  const float* arow = a0 + (size_t)nl * H_;
#pragma unroll 4
  for (int k = 0; k < H_; k += 4) {
    const int kk = k + kofs;
    v2f a = *(const v2f*)&arow[kk];
    const float* wp = Wp + (size_t)(kk >> 1) * (2 * H_);
#pragma unroll
    for (int nt = 0; nt < 8; ++nt) {
      v2f b = *(const v2f*)&wp[2 * (nt * 16 + nl)];
      acc[nt] = wmma4(a, b, acc[nt]);
    }
  }
}

// out(16x128) = A_lds(16x128) @ Wp(packed 128x128) + bias
__device__ __forceinline__ void gemm_from_lds(const float* __restrict__ Wp,
                                              const float* __restrict__ bias,
                                              const float* lds, int lane,
                                              v8f acc[8]) {
  const int nl = lane & 15;
  const int kofs = (lane >> 4) << 1;
  bias_init(acc, bias, nl);
#pragma unroll 4
  for (int k = 0; k < H_; k += 4) {
    const int kk = k + kofs;
    v2f a = *(const v2f*)&lds[ldsIdx(nl, kk)];
    const float* wp = Wp + (size_t)(kk >> 1) * (2 * H_);
#pragma unroll
    for (int nt = 0; nt < 8; ++nt) {
      v2f b = *(const v2f*)&wp[2 * (nt * 16 + nl)];
      acc[nt] = wmma4(a, b, acc[nt]);
    }
  }
}

// apply ssp to C/D fragment and store 16x128 tile to swizzled LDS
__device__ __forceinline__ void store_ssp_to_lds(float* lds, int lane,
                                                 const v8f acc[8]) {
  const int nl = lane & 15;
  const int mo = (lane >> 4) << 3;
#pragma unroll
  for (int nt = 0; nt < 8; ++nt)
#pragma unroll
    for (int r = 0; r < 8; ++r)
      lds[ldsIdx(mo + r, nt * 16 + nl)] = sspf(acc[nt][r]);
}

// ---------------------------------------------------------------------------
// Weight pre-pack: row-pair interleave + zero padding (K 50->64 for eW1)
// ---------------------------------------------------------------------------
__global__ void k_pack(const float* __restrict__ W, float* __restrict__ Wp,
                       int Kreal, int Kpad) {
  int t = blockIdx.x * blockDim.x + threadIdx.x;
  if (t >= (Kpad >> 1) * H_) return;
  int pr = t >> 7, n = t & 127;
  int k0 = pr * 2;
  Wp[2 * t + 0] = (k0 < Kreal) ? W[k0 * H_ + n] : 0.f;
  Wp[2 * t + 1] = (k0 + 1 < Kreal) ? W[(k0 + 1) * H_ + n] : 0.f;
}

// ---------------------------------------------------------------------------
// Kernel 1: per-edge distance + soft cutoff
// ---------------------------------------------------------------------------
__global__ void k_geom(const int* __restrict__ edges, const float* __restrict__ xyz,
                       const float* __restrict__ disp, const float* __restrict__ cell,
                       float* __restrict__ dist, float* __restrict__ soft) {
  int g = blockIdx.x * blockDim.x + threadIdx.x;
  if (g >= BE_) return;
  int b = g / E_;
  const float* cb = cell + b * 9;
  float d0 = disp[3 * g + 0], d1 = disp[3 * g + 1], d2 = disp[3 * g + 2];
  float dc0 = d0 * cb[0] + d1 * cb[3] + d2 * cb[6];
  float dc1 = d0 * cb[1] + d1 * cb[4] + d2 * cb[7];
  float dc2 = d0 * cb[2] + d1 * cb[5] + d2 * cb[8];
  int s = edges[2 * g + 0] + b * N_;
  int r = edges[2 * g + 1] + b * N_;
  float dx = xyz[3 * r + 0] - (xyz[3 * s + 0] + dc0);
  float dy = xyz[3 * r + 1] - (xyz[3 * s + 1] + dc1);
  float dz = xyz[3 * r + 2] - (xyz[3 * s + 2] + dc2);
  float dd = sqrtf(dx * dx + dy * dy + dz * dz);
  dist[g] = dd;
  soft[g] = 1.f / (1.f + __expf(5.f * (dd - 3.5f)));  // 1 - sigmoid(5(d-3.5))
}

// ---------------------------------------------------------------------------
// Kernel 2: embedding gather
// ---------------------------------------------------------------------------
__global__ void k_embed(const int* __restrict__ nodes, const float* __restrict__ embed,
                        float* __restrict__ state) {
  int t = blockIdx.x * blockDim.x + threadIdx.x;
  if (t >= BN_ * H_) return;
  state[t] = embed[nodes[t >> 7] * H_ + (t & 127)];
}

__global__ void k_zero(float* __restrict__ p, int n) {
  int t = blockIdx.x * blockDim.x + threadIdx.x;
  if (t < n) p[t] = 0.f;
}

// ---------------------------------------------------------------------------
// Kernel 3: fused per-edge pipeline.
// Block = 128 threads = 4 waves; each wave owns a 16-edge tile (M=16) and
// computes gates (edge MLP) and messages (node MLP) entirely on WMMA, then
// scatters gated messages with fp32 atomics.
// ---------------------------------------------------------------------------
__global__ void __launch_bounds__(128) k_edge(
    const int* __restrict__ edges, const float* __restrict__ dist,
    const float* __restrict__ soft, const float* __restrict__ state,
    const float* __restrict__ eW1p, const float* __restrict__ eb1,
    const float* __restrict__ eW2p, const float* __restrict__ eb2,
    const float* __restrict__ nW1p, const float* __restrict__ nb1,
    const float* __restrict__ nW2p, const float* __restrict__ nb2,
    float* __restrict__ msg_sum) {
  __shared__ float sh[4 * 16 * H_];  // 32 KB: per-wave 16x128 hidden tile
  const int lane = threadIdx.x & 31;
  const int wave = threadIdx.x >> 5;
  float* lds = sh + wave * 16 * H_;
  const int base = blockIdx.x * 64 + wave * 16;  // BE_=400000 = 6250*64 exact
  const int nl = lane & 15;
  const int kofs = (lane >> 4) << 1;  // K offset of this lane half
  const int mo = (lane >> 4) << 3;    // M offset of this lane half in C/D

  // per-lane edge scalars for A-fragment row (lane&15)
  const int gr = base + nl;
  const int br = gr / E_;
  const int sNode = edges[2 * gr + 0] + br * N_;
  const int rNode = edges[2 * gr + 1] + br * N_;
  const float dr = dist[gr];

  v8f acc[8];

  // ---- edge MLP stage 1: hidden = ssp(gauss(dist) @ eW1 + eb1), K=50 pad 64
  // Padding rows of eW1p are zero, and the Gaussian is always finite, so no
  // masking is needed at all.
  bias_init(acc, eb1, nl);
#pragma unroll 2
  for (int k = 0; k < EDGE_PAD; k += 4) {
    const int j = k + kofs;
    float t0 = dr - 0.1f * (float)j;
    float t1 = t0 - 0.1f;
    v2f a;
    a.x = __expf(-50.f * t0 * t0);  // exp(-(d-mu)^2 / (2*0.1^2))
    a.y = __expf(-50.f * t1 * t1);
    const float* wp = eW1p + (size_t)(j >> 1) * (2 * H_);
#pragma unroll
    for (int nt = 0; nt < 8; ++nt) {
      v2f b = *(const v2f*)&wp[2 * (nt * 16 + nl)];
      acc[nt] = wmma4(a, b, acc[nt]);
    }
  }
  store_ssp_to_lds(lds, lane, acc);
  __syncthreads();

  // ---- edge MLP stage 2: gates = hidden @ eW2 + eb2
  v8f gacc[8];
  gemm_from_lds(eW2p, eb2, lds, lane, gacc);
  __syncthreads();

  // ---- node MLP stage 1: hidden = ssp([send||recv] @ nW1 + nb1), K=256
  // split into the two K=128 halves so the gather pointer is loop-invariant
  bias_init(acc, nb1, nl);
  gemm_from_global(state + (size_t)sNode * H_ - (size_t)nl * H_, nW1p, lane, acc);
  gemm_from_global(state + (size_t)rNode * H_ - (size_t)nl * H_,
                   nW1p + (size_t)H_ * H_, lane, acc);
  store_ssp_to_lds(lds, lane, acc);
  __syncthreads();

  // ---- node MLP stage 2: msg = hidden @ nW2 + nb2
  v8f macc[8];
  gemm_from_lds(nW2p, nb2, lds, lane, macc);

  // ---- msg *= gates * soft_cut; scatter-add into msg_sum[recv]
#pragma unroll
  for (int r = 0; r < 8; ++r) {
    const int g = base + mo + r;
    const int b = g / E_;
    const int rn = edges[2 * g + 1] + b * N_;
    const float sc = soft[g];
    float* dst = msg_sum + (size_t)rn * H_;
#pragma unroll
    for (int nt = 0; nt < 8; ++nt) {
      atomicAdd(dst + nt * 16 + nl, macc[nt][r] * gacc[nt][r] * sc);
    }
  }
}

// ---------------------------------------------------------------------------
// Kernel 4: per-node state update:
//   state += ssp(msg_sum @ sW1 + sb1) @ sW2 + sb2 ; out[layer] = state
// ---------------------------------------------------------------------------
__global__ void __launch_bounds__(128) k_state(
    const float* __restrict__ msg_sum, const float* __restrict__ sW1p,
    const float* __restrict__ sb1, const float* __restrict__ sW2p,
    const float* __restrict__ sb2, float* __restrict__ state,
    float* __restrict__ out) {
  __shared__ float sh[4 * 16 * H_];
  const int lane = threadIdx.x & 31;
  const int wave = threadIdx.x >> 5;
  float* lds = sh + wave * 16 * H_;
  const int base = blockIdx.x * 64 + wave * 16;  // BN_=16000 = 250*64 exact
  const int nl = lane & 15;
  const int mo = (lane >> 4) << 3;

  v8f acc[8];
  bias_init(acc, sb1, nl);
  gemm_from_global(msg_sum + (size_t)base * H_, sW1p, lane, acc);
  store_ssp_to_lds(lds, lane, acc);
  __syncthreads();

  v8f oacc[8];
  gemm_from_lds(sW2p, sb2, lds, lane, oacc);

#pragma unroll
  for (int r = 0; r < 8; ++r) {
    const size_t row = (size_t)(base + mo + r);
#pragma unroll
    for (int nt = 0; nt < 8; ++nt) {
      const size_t idx = row * H_ + nt * 16 + nl;
      const float v = oacc[nt][r] + state[idx];
      state[idx] = v;
      out[idx] = v;
    }
  }
}

// ---------------------------------------------------------------------------
// Host launcher.
// d_in order (jax tree leaves; layer dicts sort == insertion order):
// 0 nodes(i32) 1 xyz 2 edges(i32) 3 disp 4 cell 5 num_nodes 6 num_edges
// 7 embed, then per layer l: 8+12l .. : eW1,eb1,eW2,eb2,nW1,nb1,nW2,nb2,
//                                        sW1,sb1,sW2,sb2
// ---------------------------------------------------------------------------
extern "C" void kernel_launch(void* const* d_in, const int* in_sizes, int n_in,
                              void* d_out, int out_size, void* d_ws,
                              size_t ws_size, hipStream_t stream) {
  (void)in_sizes; (void)n_in; (void)out_size; (void)ws_size;
  const int* nodes = (const int*)d_in[0];
  const float* xyz = (const float*)d_in[1];
  const int* edges = (const int*)d_in[2];
  const float* disp = (const float*)d_in[3];
  const float* cell = (const float*)d_in[4];
  const float* embed = (const float*)d_in[7];

  float* ws = (float*)d_ws;
  float* dist = ws;                               // BE_
  float* soft = ws + BE_;                         // BE_
  float* state = ws + 2 * BE_;                    // BN_*H_
  float* msgsum = ws + 2 * BE_ + BN_ * H_;        // BN_*H_
  float* packs = ws + 2 * BE_ + 2 * BN_ * H_;     // 3 * 106496 packed weights

  // per-layer packed weight offsets (floats)
  const int PL = 106496;  // 8192+16384+32768+16384+16384+16384
  // pack all weights (eW1 padded 50->64 with zero rows baked in)
  for (int l = 0; l < 3; ++l) {
    const float* p[12];
    for (int i = 0; i < 12; ++i) p[i] = (const float*)d_in[8 + 12 * l + i];
    float* pk = packs + (size_t)l * PL;
    k_pack<<<(32 * H_ + 255) / 256, 256, 0, stream>>>(p[0], pk, EDGE_REAL, EDGE_PAD);
    k_pack<<<(64 * H_ + 255) / 256, 256, 0, stream>>>(p[2], pk + 8192, H_, H_);
    k_pack<<<(128 * H_ + 255) / 256, 256, 0, stream>>>(p[4], pk + 24576, 256, 256);
    k_pack<<<(64 * H_ + 255) / 256, 256, 0, stream>>>(p[6], pk + 57344, H_, H_);
    k_pack<<<(64 * H_ + 255) / 256, 256, 0, stream>>>(p[8], pk + 73728, H_, H_);
    k_pack<<<(64 * H_ + 255) / 256, 256, 0, stream>>>(p[10], pk + 90112, H_, H_);
  }

  k_geom<<<(BE_ + 255) / 256, 256, 0, stream>>>(edges, xyz, disp, cell, dist, soft);
  k_embed<<<(BN_ * H_ + 255) / 256, 256, 0, stream>>>(nodes, embed, state);

  for (int l = 0; l < 3; ++l) {
    const float* p[12];
    for (int i = 0; i < 12; ++i) p[i] = (const float*)d_in[8 + 12 * l + i];
    float* pk = packs + (size_t)l * PL;
    k_zero<<<(BN_ * H_ + 255) / 256, 256, 0, stream>>>(msgsum, BN_ * H_);
    k_edge<<<BE_ / 64, 128, 0, stream>>>(edges, dist, soft, state,
                                         pk, p[1], pk + 8192, p[3],
                                         pk + 24576, p[5], pk + 57344, p[7],
                                         msgsum);
    k_state<<<BN_ / 64, 128, 0, stream>>>(msgsum, pk + 73728, p[9],
                                          pk + 90112, p[11], state,
                                          (float*)d_out + (size_t)l * BN_ * H_);
  }
}